// Net_55989193670682
// MI455X (gfx1250) — compile-verified
//
#include <hip/hip_runtime.h>

typedef float v2f __attribute__((ext_vector_type(2)));
typedef float v8f __attribute__((ext_vector_type(8)));

#define B_    128
#define N_    116   // nodes = K of GEMM1 (= 29 * 4, exact for 16x16x4 WMMA)
#define E_    670   // edges
#define INV_  116
#define INE_  5
#define OUTE_ 64
#define ETILES 42          // ceil(670/16)
#define EPAD   672         // 42*16
#define NA_STRIDE 676      // lane*676 mod 64 = lane*36 mod 64 -> 16 distinct banks,
                           // even -> 8B-aligned v2f reads, lo/hi b64 pairs disjoint
#define HE_STRIDE 8        // He rows padded to 8 floats -> b128+b32 reads, 32B aligned

// One workgroup = 4 wave32 = 128 threads, owns rows [e0, e0+16) of one batch.
__global__ __launch_bounds__(128) void edge_update_fused(
    const float* __restrict__ Hv, const float* __restrict__ He,
    const float* __restrict__ Ae, const float* __restrict__ T,
    const float* __restrict__ Wt, const float* __restrict__ p,
    const float* __restrict__ bias, float* __restrict__ out)
{
    __shared__ __align__(16) float sA[16 * N_];            // (T col-slice)*d    7.4 KB
    __shared__ __align__(16) float sB[4][16 * N_];         // B tile per wave   29.7 KB
    __shared__ __align__(16) float sNA[16 * NA_STRIDE];    // normA stripe      43.3 KB
    __shared__ __align__(16) float sHe[EPAD * HE_STRIDE];  // He[b], padded     21.5 KB
    __shared__ float sd[N_];
    __shared__ float sPart[128];
    __shared__ float sInv[16];

    const int b    = blockIdx.x / ETILES;
    const int e0   = (blockIdx.x % ETILES) * 16;
    const int tid  = threadIdx.x;
    const int lane = tid & 31;                // wave32
    const int w    = tid >> 5;                // wave id 0..3
    const int lo   = lane & 15;
    const int hi   = lane >> 4;               // 0 or 1

    // ---- stage He[b] (row-padded to 8) and compute d = Hv @ p ----
    {
        const float* heb = He + (size_t)b * E_ * INE_;
        for (int i = tid; i < EPAD * HE_STRIDE; i += 128) {
            const int k = i >> 3, f = i & 7;
            sHe[i] = (f < INE_ && k < E_) ? heb[(size_t)k * INE_ + f] : 0.0f;
        }
        if (tid < N_) {
            const float* hvr = Hv + ((size_t)b * N_ + tid) * INV_;
            float acc = 0.0f;
            for (int f = 0; f < INV_; ++f) acc += hvr[f] * p[f];
            sd[tid] = acc;
        }
    }
    __syncthreads();

    // ---- stage A rows: sA[i][n] = T[b, n, e0+i] * d[n]  (clamped, no divergence) ----
    {
        const float* Tb = T + (size_t)b * N_ * E_;
        for (int idx = tid; idx < 16 * N_; idx += 128) {
            const int i = idx & 15, n = idx >> 4;
            const int e  = e0 + i;
            const int ec = (e < E_) ? e : (E_ - 1);
            const float t = Tb[(size_t)n * E_ + ec];        // unconditional load
            sA[i * N_ + n] = (e < E_) ? t * sd[n] : 0.0f;   // value select only
        }
    }
    __syncthreads();

    // ---- phase 1: mult2 tiles = A @ B via f32 WMMA; fuse diag/adj_e -> sNA ----
    {
        const float* Tb  = T  + (size_t)b * N_ * E_;
        const float* Aeb = Ae + (size_t)b * E_ * E_;
        float* sBw = sB[w];
        const int r0c = ((e0 + lo) < E_) ? (e0 + lo) : (E_ - 1);
        for (int ft = w; ft < ETILES; ft += 4) {
            const int f0  = ft * 16;
            const int col = f0 + lo;                         // global column
            const bool cok = (col < E_);
            __builtin_prefetch(&Aeb[(size_t)r0c * E_ + f0], 0, 3);

            // wave-cooperative B-tile staging: sBw[i][n] = T[b, n, f0+i] (clamped)
            for (int idx = lane; idx < 16 * N_; idx += 32) {
                const int i = idx & 15, n = idx >> 4;
                const int e  = f0 + i;
                const int ec = (e < E_) ? e : (E_ - 1);      // garbage cols zeroed later
                sBw[i * N_ + n] = Tb[(size_t)n * E_ + ec];
            }

            v8f c = {};
            #pragma unroll 4
            for (int k0 = 0; k0 < N_; k0 += 4) {
                const int ka = k0 + 2 * hi;
                const v2f av = *(const v2f*)&sA[lo * N_ + ka];   // A[M=lo][K=ka..ka+1]
                const v2f bv = *(const v2f*)&sBw[lo * N_ + ka];  // B[K=ka..ka+1][N=lo]
                c = __builtin_amdgcn_wmma_f32_16x16x4_f32(
                        false, av, false, bv, (short)0, c, false, false);
            }
            const int colc = cok ? col : (E_ - 1);
            #pragma unroll
            for (int j = 0; j < 8; ++j) {
                const int lr  = j + 8 * hi;                  // local row 0..15
                const int row = e0 + lr;                     // global row
                const int rc  = (row < E_) ? row : (E_ - 1);
                const float ae = Aeb[(size_t)rc * E_ + colc]; // unconditional, coalesced
                const bool valid = (row < E_) && cok;
                const float v = (row == col) ? 1.0f : c[j];
                sNA[lr * NA_STRIDE + col] = valid ? v * ae : 0.0f;
            }
        }
    }
    __syncthreads();

    // ---- per-row max over real columns, then scale by reciprocal ----
    {
        const int row = tid >> 3, sub = tid & 7;
        float m = -3.4e38f;
        for (int cI = sub; cI < E_; cI += 8)
            m = fmaxf(m, sNA[row * NA_STRIDE + cI]);
        sPart[tid] = m;
    }
    __syncthreads();
    if (tid < 16) {
        float m = sPart[tid * 8];
        #pragma unroll
        for (int s = 1; s < 8; ++s) m = fmaxf(m, sPart[tid * 8 + s]);
        sInv[tid] = 1.0f / (m + 1e-10f);
    }
    __syncthreads();
    for (int idx = tid; idx < 16 * EPAD; idx += 128) {
        const int row = idx / EPAD, cI = idx - row * EPAD;
        sNA[row * NA_STRIDE + cI] *= sInv[row];
    }
    __syncthreads();

    // ---- phase 2: out[16 x 64] = normA(16 x 672) @ (He @ W), B built on the fly ----
    {
        const int gc = w * 16 + lo;              // output column 0..63
        const float W0 = Wt[0 * OUTE_ + gc];
        const float W1 = Wt[1 * OUTE_ + gc];
        const float W2 = Wt[2 * OUTE_ + gc];
        const float W3 = Wt[3 * OUTE_ + gc];
        const float W4 = Wt[4 * OUTE_ + gc];
        const float bs = bias[gc];
        v8f c = {};
        #pragma unroll 4
        for (int k0 = 0; k0 < EPAD; k0 += 4) {
            const int ka = k0 + 2 * hi;
            const v2f av = *(const v2f*)&sNA[lo * NA_STRIDE + ka];  // ds_load_b64
            const float4 h0 = *(const float4*)&sHe[ka * HE_STRIDE];         // b128
            const float  h04 = sHe[ka * HE_STRIDE + 4];                     // b32
            const float4 h1 = *(const float4*)&sHe[(ka + 1) * HE_STRIDE];
            const float  h14 = sHe[(ka + 1) * HE_STRIDE + 4];
            v2f bb;
            bb.x = h0.x*W0 + h0.y*W1 + h0.z*W2 + h0.w*W3 + h04*W4;
            bb.y = h1.x*W0 + h1.y*W1 + h1.z*W2 + h1.w*W3 + h14*W4;
            c = __builtin_amdgcn_wmma_f32_16x16x4_f32(
                    false, av, false, bb, (short)0, c, false, false);
        }
        float* ob = out + (size_t)b * E_ * OUTE_;
        #pragma unroll
        for (int j = 0; j < 8; ++j) {
            const int row = e0 + j + 8 * hi;
            if (row < E_) ob[(size_t)row * OUTE_ + gc] = c[j] + bs;
        }
    }
}

// Pass-through of Hv (first tuple output), vectorized copy.
__global__ void copy_hv_kernel(const float* __restrict__ src,
                               float* __restrict__ dst, int n4)
{
    int i = blockIdx.x * blockDim.x + threadIdx.x;
    if (i < n4) ((float4*)dst)[i] = ((const float4*)src)[i];
}

extern "C" void kernel_launch(void* const* d_in, const int* in_sizes, int n_in,
                              void* d_out, int out_size, void* d_ws, size_t ws_size,
                              hipStream_t stream) {
    const float* Hv   = (const float*)d_in[0];
    const float* He   = (const float*)d_in[1];
    const float* Ae   = (const float*)d_in[2];
    // d_in[3] = adj_v : unused by the edge-update layer
    const float* T    = (const float*)d_in[4];
    const float* Wt   = (const float*)d_in[5];
    const float* p    = (const float*)d_in[6];
    const float* bias = (const float*)d_in[7];

    float* out_hv   = (float*)d_out;
    float* out_edge = out_hv + (size_t)B_ * N_ * INV_;   // tuple: (Hv, out)

    const int nhv4 = (B_ * N_ * INV_) / 4;               // 1,722,368 / 4
    copy_hv_kernel<<<(nhv4 + 255) / 256, 256, 0, stream>>>(Hv, out_hv, nhv4);

    edge_update_fused<<<B_ * ETILES, 128, 0, stream>>>(
        Hv, He, Ae, T, Wt, p, bias, out_edge);
}